// Semantic_Mapping_87789131530939
// MI455X (gfx1250) — compile-verified
//
#include <hip/hip_runtime.h>
#include <hip/hip_bf16.h>
#include <stdint.h>

// ---------------------------------------------------------------------------
// Semantic mapping forward, fused for MI455X (gfx1250, wave32).
//  K1: zero 100x100 accumulator in workspace
//  K2: depth -> point -> z-collapsed trilinear splat, LDS ds_add_f32 atomics,
//      per-block merge via global atomics
//  K3: async-DMA the 40KB map into LDS (ASYNCcnt path), fused double-bilinear
//      resample (rotation o translation composed exactly), max + threshold
// ---------------------------------------------------------------------------

namespace {
constexpr int   kMapSz  = 960;
constexpr int   kVR     = 100;
constexpr int   kCells  = kVR * kVR;          // 10000 floats = 40 KB
constexpr int   kNPix   = 1024 * 1024;
constexpr float kInvFoc = 1.0f / 512.0f;      // FOC = 512
}

// gfx1250 async global->LDS path (ASYNCcnt), guarded: fall back to plain copy
#if defined(__gfx1250__) && __has_builtin(__builtin_amdgcn_global_load_async_to_lds_b128) && __has_builtin(__builtin_amdgcn_s_wait_asynccnt)
#define SM_ASYNC_B128 1
#elif defined(__gfx1250__) && __has_builtin(__builtin_amdgcn_global_load_async_to_lds_b32) && __has_builtin(__builtin_amdgcn_s_wait_asynccnt)
#define SM_ASYNC_B32 1
#endif

// Builtin prototypes take (v4i as1*, v4i as3*, imm, imm) for b128 and
// (int as1*, int as3*, imm, imm) for b32 (per hipcc diagnostic).
typedef int sm_v4i __attribute__((ext_vector_type(4)));
typedef __attribute__((address_space(1))) sm_v4i as1_v4i;
typedef __attribute__((address_space(3))) sm_v4i as3_v4i;
typedef __attribute__((address_space(1))) int    as1_i32;
typedef __attribute__((address_space(3))) int    as3_i32;

// Integer round-trip avoids generic->AS cast diagnostics. On amdgcn the low
// 32 bits of a generic LDS address are the LDS byte offset; global generic
// addresses equal as(1) addresses.
static __device__ __forceinline__ as1_v4i* sm_global_v4(const void* p) {
  return (as1_v4i*)(unsigned long long)(uintptr_t)p;
}
static __device__ __forceinline__ as3_v4i* sm_lds_v4(void* p) {
  return (as3_v4i*)(unsigned)(uintptr_t)p;
}
static __device__ __forceinline__ as1_i32* sm_global_i32(const void* p) {
  return (as1_i32*)(unsigned long long)(uintptr_t)p;
}
static __device__ __forceinline__ as3_i32* sm_lds_i32(void* p) {
  return (as3_i32*)(unsigned)(uintptr_t)p;
}

// --------------------------- K1: zero accumulator ---------------------------
__global__ void sm_zero_kernel(float* __restrict__ fpacc) {
  int i = blockIdx.x * blockDim.x + threadIdx.x;
  if (i < kCells) fpacc[i] = 0.0f;
}

// --------------------------- K2: splat ---------------------------
// pos_x = (X + 250)/5 ; pos_y = depth_cm/5 ; pos_z = (Z + 88)/5 + 30
// fp contribution at cell (x,y): feat0 * wx * wy * sum_{z in [34,56]} wz
__global__ void sm_splat_kernel(const float* __restrict__ depth,
                                const float* __restrict__ feat0,
                                float* __restrict__ fpacc) {
  __shared__ float acc[kCells];
  for (int i = threadIdx.x; i < kCells; i += blockDim.x) acc[i] = 0.0f;
  __syncthreads();

  const int stride = gridDim.x * blockDim.x;
  for (int n = blockIdx.x * blockDim.x + threadIdx.x; n < kNPix; n += stride) {
    // CDNA5 speculative prefetch of next strided element (global_prefetch_b8)
    int np = (n + stride < kNPix) ? (n + stride) : n;
    __builtin_prefetch(&depth[np], 0, 1);
    __builtin_prefetch(&feat0[np], 0, 1);

    float dcm = depth[n] * 100.0f;
    int row = n >> 10;
    int col = n & 1023;
    float X  = ((float)col - 511.5f) * dcm * kInvFoc;
    float Zc = ((float)(1023 - row) - 511.5f) * dcm * kInvFoc;
    float posx = (X + 250.0f) * 0.2f;
    float posy = dcm * 0.2f;
    float posz = (Zc + 88.0f) * 0.2f + 30.0f;

    // collapse z: weights of the two z-corners that land in [MIN_Z, MAX_Z)
    float z0 = floorf(posz);
    float zsum = 0.0f;
#pragma unroll
    for (int k = 0; k < 2; ++k) {
      float zk = z0 + (float)k;
      if (zk >= 34.0f && zk <= 56.0f) zsum += 1.0f - fabsf(posz - zk);
    }
    if (zsum <= 0.0f) continue;
    float w = feat0[n] * zsum;

    float xf = floorf(posx);
    float yf = floorf(posy);
#pragma unroll
    for (int i2 = 0; i2 < 2; ++i2) {
      float xi = xf + (float)i2;
      if (!(xi > 0.0f && xi < 100.0f)) continue;          // reference safe-mask
      float wx = (1.0f - fabsf(posx - xi)) * w;
      int xbase = (int)xi * kVR;
#pragma unroll
      for (int j = 0; j < 2; ++j) {
        float yi = yf + (float)j;
        if (!(yi > 0.0f && yi < 100.0f)) continue;
        float wxy = wx * (1.0f - fabsf(posy - yi));
        if (wxy != 0.0f) atomicAdd(&acc[xbase + (int)yi], wxy);  // ds_add_f32
      }
    }
  }
  __syncthreads();
  for (int i = threadIdx.x; i < kCells; i += blockDim.x) {
    float v = acc[i];
    if (v != 0.0f) atomicAdd(&fpacc[i], v);               // global_atomic_add_f32
  }
}

// --------------------------- K3: fused resample ---------------------------
// fp is the clamped 100x100 map in [x*100+y] layout (voxel transpose folded in):
// agent_view[r][c] = fp[(c-430)*100 + (r-480)]
static __device__ __forceinline__ float sm_fp_at(const float* __restrict__ fp,
                                                 int jx, int jy) {
  if (jy < 480 || jy >= 580 || jx < 430 || jx >= 530) return 0.0f;
  return fp[(jx - 430) * kVR + (jy - 480)];
}

// rotated[iy][ix] recomputed exactly as the reference materializes it
static __device__ __forceinline__ float sm_rotated_at(const float* __restrict__ fp,
                                                      int ix, int iy,
                                                      float c, float s) {
  float Xn = (2.0f * (float)ix + 1.0f) * (1.0f / 960.0f) - 1.0f;
  float Yn = (2.0f * (float)iy + 1.0f) * (1.0f / 960.0f) - 1.0f;
  float ax = (c * Xn - s * Yn + 1.0f) * 0.5f * 959.0f;
  float ay = (s * Xn + c * Yn + 1.0f) * 0.5f * 959.0f;
  float x0f = floorf(ax), y0f = floorf(ay);
  float fx = ax - x0f, fy = ay - y0f;
  int x0 = (int)x0f, y0 = (int)y0f;
  float v;
  v  = (1.0f - fx) * (1.0f - fy) * sm_fp_at(fp, x0,     y0);
  v += fx          * (1.0f - fy) * sm_fp_at(fp, x0 + 1, y0);
  v += (1.0f - fx) * fy          * sm_fp_at(fp, x0,     y0 + 1);
  v += fx          * fy          * sm_fp_at(fp, x0 + 1, y0 + 1);
  return v;   // region test subsumes the [0,959] validity mask
}

__global__ void sm_resample_kernel(const float* __restrict__ fpacc,
                                   const float* __restrict__ maps_last,
                                   const float* __restrict__ pose,
                                   float* __restrict__ out) {
  __shared__ __attribute__((aligned(16))) float fp[kCells];

#if defined(SM_ASYNC_B128)
  {
    as1_v4i* g = sm_global_v4(fpacc);
    as3_v4i* l = sm_lds_v4(fp);
    for (int i = threadIdx.x; i < kCells / 4; i += blockDim.x)
      __builtin_amdgcn_global_load_async_to_lds_b128(g + i, l + i, 0, 0);
    __builtin_amdgcn_s_wait_asynccnt(0);
  }
#elif defined(SM_ASYNC_B32)
  {
    as1_i32* g = sm_global_i32(fpacc);
    as3_i32* l = sm_lds_i32(fp);
    for (int i = threadIdx.x; i < kCells; i += blockDim.x)
      __builtin_amdgcn_global_load_async_to_lds_b32(g + i, l + i, 0, 0);
    __builtin_amdgcn_s_wait_asynccnt(0);
  }
#else
  for (int i = threadIdx.x; i < kCells; i += blockDim.x) fp[i] = fpacc[i];
#endif
  __syncthreads();
  // fp_map = clip(sum/10, 0, 1), done once in LDS
  for (int i = threadIdx.x; i < kCells; i += blockDim.x)
    fp[i] = fminf(fmaxf(fp[i] * 0.1f, 0.0f), 1.0f);
  __syncthreads();

  // pose-derived affine params (st_pose)
  float tx = -(pose[0] * 20.0f - 480.0f) * (1.0f / 480.0f);
  float ty = -(pose[1] * 20.0f - 480.0f) * (1.0f / 480.0f);
  float th = (90.0f - pose[2]) * 0.017453292519943295f;
  float c = cosf(th);
  float s = sinf(th);

  const int total = kMapSz * kMapSz;
  const int stride = gridDim.x * blockDim.x;
  for (int p = blockIdx.x * blockDim.x + threadIdx.x; p < total; p += stride) {
    int r  = p / kMapSz;
    int cc = p - r * kMapSz;
    // translation grid (theta2), align_corners=True pixel coords
    float Xn = (2.0f * (float)cc + 1.0f) * (1.0f / 960.0f) - 1.0f;
    float Yn = (2.0f * (float)r  + 1.0f) * (1.0f / 960.0f) - 1.0f;
    float gx = (Xn + tx + 1.0f) * 0.5f * 959.0f;
    float gy = (Yn + ty + 1.0f) * 0.5f * 959.0f;
    float x0f = floorf(gx), y0f = floorf(gy);
    float fx = gx - x0f, fy = gy - y0f;
    int x0 = (int)x0f, y0 = (int)y0f;

    float accv = 0.0f;
#pragma unroll
    for (int j = 0; j < 2; ++j) {
      int iy = y0 + j;
      if (iy < 0 || iy > kMapSz - 1) continue;
      float wy = j ? fy : (1.0f - fy);
#pragma unroll
      for (int i2 = 0; i2 < 2; ++i2) {
        int ix = x0 + i2;
        if (ix < 0 || ix > kMapSz - 1) continue;
        float wxy = wy * (i2 ? fx : (1.0f - fx));
        if (wxy != 0.0f) accv += wxy * sm_rotated_at(fp, ix, iy, c, s);
      }
    }
    float v = fmaxf(maps_last[p], accv);
    out[p] = (v > 0.5f) ? 1.0f : v;
  }
}

// --------------------------- launcher ---------------------------
extern "C" void kernel_launch(void* const* d_in, const int* in_sizes, int n_in,
                              void* d_out, int out_size, void* d_ws, size_t ws_size,
                              hipStream_t stream) {
  (void)in_sizes; (void)n_in; (void)out_size; (void)ws_size;
  const float* depth     = (const float*)d_in[0];  // [1024,1024]
  const float* pose      = (const float*)d_in[1];  // [3]
  const float* maps_last = (const float*)d_in[2];  // [1,1,960,960]
  const float* feat      = (const float*)d_in[3];  // [8, 1024*1024]; only ch0 used
  float* out   = (float*)d_out;                    // [1,1,960,960]
  float* fpacc = (float*)d_ws;                     // 100*100 floats scratch

  sm_zero_kernel<<<(kCells + 255) / 256, 256, 0, stream>>>(fpacc);
  sm_splat_kernel<<<128, 256, 0, stream>>>(depth, feat, fpacc);
  sm_resample_kernel<<<900, 256, 0, stream>>>(fpacc, maps_last, pose, out);
}